// EPAll2AllLayer_54666343743635
// MI455X (gfx1250) — compile-verified
//
#include <hip/hip_runtime.h>
#include <hip/hip_bf16.h>
#include <stdint.h>
#include <stddef.h>

// MoE all-to-all dispatch preprocessing for MI455X (gfx1250).
// T=8192 tokens, H=2048, K=8, 64 experts. Pure data movement:
//   64 MB token reads + 512 MB expert-sorted writes  =>  ~25 us floor @ 23.3 TB/s.
// Heavy copy uses the CDNA5 async global<->LDS engine (ASYNCcnt) in a
// source-major broadcast: each token row is read from HBM exactly once into
// LDS, then multicast to its 8 destination rows with async B128 stores.

#define T_TOKENS 8192
#define HDIM     2048
#define KTOP     8
#define NEXPERTS 64
#define NROUTED  (T_TOKENS * KTOP)     /* 65536 routed copies            */
#define RANK_BLK 256                   /* routed entries per rank block  */
#define NBLK     (NROUTED / RANK_BLK)  /* 256 rank blocks                */
#define ROW_F4   (HDIM / 4)            /* 512 float4 per token row       */

// ---- CDNA5 async global<->LDS path (guarded so either toolchain compiles) --
#if defined(__AMDGCN__) &&                                                    \
    __has_builtin(__builtin_amdgcn_global_load_async_to_lds_b128) &&          \
    __has_builtin(__builtin_amdgcn_global_store_async_from_lds_b128)
#define USE_ASYNC_LDS 1
#else
#define USE_ASYNC_LDS 0
#endif

#if USE_ASYNC_LDS
#define AS1 __attribute__((address_space(1)))
#define AS3 __attribute__((address_space(3)))
// Builtin signature (verified via hipcc diagnostics): pointee is
// int __attribute__((vector_size(16))); param1 is the AS1 global pointer,
// param2 the AS3 LDS pointer, then imm offset and imm cpol.
typedef int v4i __attribute__((vector_size(16)));
#define GPTR(p) ((AS1 v4i*)(p))
#define LPTR(p) ((AS3 v4i*)(p))
// gfx12+ cpol immediate: TH in bits[2:0]; TH_NT = 1 (non-temporal).
#define CPOL_NT 1

__device__ __forceinline__ void async_wait0() {
#if __has_builtin(__builtin_amdgcn_s_wait_asynccnt)
  __builtin_amdgcn_s_wait_asynccnt(0);
#else
  asm volatile("s_wait_asynccnt 0" ::: "memory");
#endif
}
#endif

// ---------------------------------------------------------------------------
// K4 (defined first so the disasm snippet shows it): the bandwidth kernel.
// One block (8 waves) per source token. Stage the 8 KB row into LDS with
// async B128 loads (HBM read exactly once), then multicast to the 8
// destination rows with async B128 stores from LDS. Each thread stores only
// the LDS bytes it loaded itself -> a per-wave s_wait_asynccnt 0 between the
// phases is the only sync needed; no barrier. NT temporal hint: the 512 MB
// output is write-once and the 64 MB input is read-once, so neither should
// occupy L2.
// ---------------------------------------------------------------------------
__global__ void __launch_bounds__(256)
k_dispatch(const float* __restrict__ tokens, const int* __restrict__ scat,
           float* __restrict__ out_tokens) {
  const int t   = threadIdx.x;
  const int tok = blockIdx.x;
  const float4* __restrict__ src =
      (const float4*)tokens + (size_t)tok * ROW_F4;

#if USE_ASYNC_LDS
  __shared__ float4 buf[ROW_F4];  // 8 KB staging
  // issue 2 async b128 loads per thread: global -> LDS
  __builtin_amdgcn_global_load_async_to_lds_b128(
      GPTR(src + t), LPTR(&buf[t]), 0, CPOL_NT);
  __builtin_amdgcn_global_load_async_to_lds_b128(
      GPTR(src + t + 256), LPTR(&buf[t + 256]), 0, CPOL_NT);
  // overlap the destination-index loads with the async staging
  int dsts[KTOP];
#pragma unroll
  for (int k = 0; k < KTOP; ++k) dsts[k] = scat[tok * KTOP + k];
  async_wait0();  // LDS staging for this wave's lanes complete
#pragma unroll
  for (int k = 0; k < KTOP; ++k) {
    float4* __restrict__ drow =
        (float4*)out_tokens + (size_t)dsts[k] * ROW_F4;
    __builtin_amdgcn_global_store_async_from_lds_b128(
        GPTR(drow + t), LPTR(&buf[t]), 0, CPOL_NT);
    __builtin_amdgcn_global_store_async_from_lds_b128(
        GPTR(drow + t + 256), LPTR(&buf[t + 256]), 0, CPOL_NT);
  }
  async_wait0();  // drain before wave exit
#else
  // fallback: register-staged b128 copy (still reads HBM once, L2 serves rest)
  const float4 a = src[t];
  const float4 b = src[t + 256];
#pragma unroll
  for (int k = 0; k < KTOP; ++k) {
    const int dst = scat[tok * KTOP + k];
    float4* __restrict__ drow = (float4*)out_tokens + (size_t)dst * ROW_F4;
    drow[t]       = a;
    drow[t + 256] = b;
  }
#endif
}

// ---------------------------------------------------------------------------
// K1: per-block expert histogram (LDS ds_add atomics). 256 blocks x 256 thr.
// ---------------------------------------------------------------------------
__global__ void __launch_bounds__(RANK_BLK)
k_hist(const int* __restrict__ eidx, int* __restrict__ blockCounts) {
  __shared__ int h[NEXPERTS];
  const int t = threadIdx.x;
  if (t < NEXPERTS) h[t] = 0;
  __syncthreads();
  const int e = eidx[blockIdx.x * RANK_BLK + t];
  atomicAdd(&h[e], 1);
  __syncthreads();
  if (t < NEXPERTS) blockCounts[blockIdx.x * NEXPERTS + t] = h[t];
}

// ---------------------------------------------------------------------------
// K2: single block, 64 threads (one per expert).
//   - exclusive prefix over blocks per expert  -> blockBase (pre expert base)
//   - expert totals -> splits output, exclusive scan over experts
//   - add expert base into blockBase
// ---------------------------------------------------------------------------
__global__ void __launch_bounds__(NEXPERTS)
k_scan(const int* __restrict__ blockCounts, int* __restrict__ blockBase,
       float* __restrict__ splits_out) {
  __shared__ int totals[NEXPERTS];
  __shared__ int base[NEXPERTS];
  const int e = threadIdx.x;  // 0..63
  int run = 0;
  for (int b = 0; b < NBLK; ++b) {
    blockBase[b * NEXPERTS + e] = run;          // within-expert prefix
    run += blockCounts[b * NEXPERTS + e];
  }
  totals[e] = run;
  __syncthreads();
  if (e == 0) {
    int acc = 0;
    for (int i = 0; i < NEXPERTS; ++i) { base[i] = acc; acc += totals[i]; }
  }
  __syncthreads();
  const int eb = base[e];
  for (int b = 0; b < NBLK; ++b) blockBase[b * NEXPERTS + e] += eb;
  splits_out[e] = (float)run;                   // splits (int -> f32 buffer)
}

// ---------------------------------------------------------------------------
// K3: stable rank inside each 256-entry chunk + scatter metadata.
//   dst(i) = blockBase[blk][e] + #{j < i in chunk : e_j == e_i}
// Writes scatter_idx (float out + int ws copy) and scatters out_weights.
// ---------------------------------------------------------------------------
__global__ void __launch_bounds__(RANK_BLK)
k_rank(const int* __restrict__ eidx, const float* __restrict__ wflat,
       const int* __restrict__ blockBase, int* __restrict__ scatter_i32,
       float* __restrict__ out_w, float* __restrict__ out_scatter) {
  __shared__ int le[RANK_BLK];
  const int t = threadIdx.x;
  const int i = blockIdx.x * RANK_BLK + t;
  const int e = eidx[i];
  le[t] = e;
  __syncthreads();
  int rank = 0;
  for (int j = 0; j < t; ++j) rank += (le[j] == e) ? 1 : 0;
  const int dst = blockBase[blockIdx.x * NEXPERTS + e] + rank;
  scatter_i32[i] = dst;
  out_scatter[i] = (float)dst;
  out_w[dst]     = wflat[i];
}

// ---------------------------------------------------------------------------
extern "C" void kernel_launch(void* const* d_in, const int* in_sizes, int n_in,
                              void* d_out, int out_size, void* d_ws,
                              size_t ws_size, hipStream_t stream) {
  (void)in_sizes; (void)n_in; (void)out_size; (void)ws_size;

  const float* tokens = (const float*)d_in[0];  // [T, H] f32
  const int*   eidx   = (const int*)d_in[1];    // [T, K] i32
  const float* weight = (const float*)d_in[2];  // [T, K] f32

  // d_out layout (all float32): out_tokens | out_weights | splits | scatter
  float* out_tokens  = (float*)d_out;
  float* out_weights = out_tokens + (size_t)NROUTED * HDIM;
  float* out_splits  = out_weights + NROUTED;
  float* out_scatter = out_splits + NEXPERTS;

  // workspace layout
  char* ws          = (char*)d_ws;
  int*  blockCounts = (int*)(ws);                                // 64 KB
  int*  blockBase   = (int*)(ws + (size_t)NBLK * NEXPERTS * 4);  // 64 KB
  int*  scatter_i32 = (int*)(ws + (size_t)2 * NBLK * NEXPERTS * 4);  // 256 KB

  k_hist<<<NBLK, RANK_BLK, 0, stream>>>(eidx, blockCounts);
  k_scan<<<1, NEXPERTS, 0, stream>>>(blockCounts, blockBase, out_splits);
  k_rank<<<NBLK, RANK_BLK, 0, stream>>>(eidx, weight, blockBase, scatter_i32,
                                        out_weights, out_scatter);
  k_dispatch<<<T_TOKENS, 256, 0, stream>>>(tokens, scatter_i32, out_tokens);
}